// ALSRLoss_82240033784032
// MI455X (gfx1250) — compile-verified
//
#include <hip/hip_runtime.h>
#include <math.h>

#define NUM_CLASSES 75000
#define BATCH_SZ    512
#define NCH         10            // chunks per row: 75000/10 = 7500 floats = 1875 float4 (16B aligned)
#define CHUNK_V4    1875
#define EPS_C       0.1f
#define ALPHA_C     0.2f

typedef __attribute__((ext_vector_type(2))) float v2f;
typedef __attribute__((ext_vector_type(8))) float v8f;

// online-softmax accumulator update: t += v; (m,s) <- merge exp stats
__device__ __forceinline__ void onl_update(float v, float& m, float& s, float& t) {
    t += v;
    if (v > m) { s = s * __expf(m - v) + 1.0f; m = v; }
    else       { s += __expf(v - m); }
}

__device__ __forceinline__ void onl_merge(float om, float os, float ot,
                                          float& m, float& s, float& t) {
    float mn = fmaxf(m, om);
    s = s * __expf(m - mn) + os * __expf(om - mn);
    m = mn;
    t += ot;
}

// Pass 1: per (row, chunk) compute (max, sum exp(x-max), sum x)
__global__ __launch_bounds__(256)
void alsr_partials(const float* __restrict__ x, float* __restrict__ part) {
    const int r   = blockIdx.x;   // 0..511
    const int c   = blockIdx.y;   // 0..NCH-1
    const int tid = threadIdx.x;

    const float4* base = (const float4*)(x + (long long)r * NUM_CLASSES) + c * CHUNK_V4;

    float m = -INFINITY, s = 0.0f, t = 0.0f;
    for (int i = tid; i < CHUNK_V4; i += 256) {
        __builtin_prefetch(base + i + 256, 0, 0);   // global_prefetch_b8
        float4 q = base[i];
        onl_update(q.x, m, s, t);
        onl_update(q.y, m, s, t);
        onl_update(q.z, m, s, t);
        onl_update(q.w, m, s, t);
    }

    // wave32 reduction
    #pragma unroll
    for (int off = 16; off > 0; off >>= 1) {
        float om = __shfl_xor(m, off, 32);
        float os = __shfl_xor(s, off, 32);
        float ot = __shfl_xor(t, off, 32);
        onl_merge(om, os, ot, m, s, t);
    }

    __shared__ float mlds[8], slds[8], tlds[8];
    const int wave = tid >> 5, lane = tid & 31;
    if (lane == 0) { mlds[wave] = m; slds[wave] = s; tlds[wave] = t; }
    __syncthreads();

    if (tid == 0) {
        float mm = mlds[0], ss = slds[0], tt = tlds[0];
        #pragma unroll
        for (int w = 1; w < 8; ++w) onl_merge(mlds[w], slds[w], tlds[w], mm, ss, tt);
        float* p = part + (r * NCH + c) * 3;
        p[0] = mm; p[1] = ss; p[2] = tt;
    }
}

// Pass 2: per-row closed-form loss, then WMMA fp32 reduction of 512 values.
__global__ __launch_bounds__(512)
void alsr_finalize(const float* __restrict__ x,
                   const int* __restrict__ pids,
                   const int* __restrict__ vids,
                   const float* __restrict__ part,
                   float* __restrict__ out) {
    __shared__ float lds[BATCH_SZ];
    const int r = threadIdx.x;    // one thread per row

    // merge the NCH chunk partials
    float m = -INFINITY, s = 0.0f, t = 0.0f;
    #pragma unroll
    for (int c = 0; c < NCH; ++c) {
        const float* p = part + (r * NCH + c) * 3;
        onl_merge(p[0], p[1], p[2], m, s, t);
    }
    const float lse = m + logf(s);

    const float* row = x + (long long)r * NUM_CLASSES;
    const int base = pids[r] * 3;
    const int v    = vids[r];
    const float x0 = row[base], x1 = row[base + 1], x2 = row[base + 2];
    const float xv = (v == 0) ? x0 : (v == 1 ? x1 : x2);

    const float p0 = __expf(x0 - lse), p1 = __expf(x1 - lse), p2 = __expf(x2 - lse);
    const float pv = (v == 0) ? p0 : (v == 1 ? p1 : p2);

    const float ep1 = ALPHA_C * (1.0f - (p0 + p1 + p2));
    const float ep2 = ALPHA_C * (1.0f - pv);
    const float eK  = EPS_C / (float)NUM_CLASSES;
    const float c0  = (1.0f - EPS_C) * ep1 / (float)(NUM_CLASSES - 3) + eK;
    const float c1  = (1.0f - EPS_C) * 0.5f * ep2 + eK;
    const float c2  = (1.0f - EPS_C) * (1.0f - ep1 - ep2) + eK;

    const float dot = c0 * t + (c1 - c0) * (x0 + x1 + x2) + (c2 - c1) * xv;
    lds[r] = (lse - dot) * (1.0f / (float)BATCH_SZ);
    __syncthreads();

    // Final 512 -> 1 reduction on the matrix unit: D = A x ones + C, fp32.
    // A 16x4 layout (ISA 7.12.2): VGPR0 = K0 (lanes 0-15, M=lane) / K2 (lanes 16-31),
    // VGPR1 = K1 / K3.  Row m of D therefore sums lds[i*64 + {0,16,32,48} + m],
    // i.e. all indices congruent to m (mod 16): 8 accumulating WMMAs cover all 512.
    if (threadIdx.x < 32) {
        const int lane = threadIdx.x;
        v8f acc = {};
        v2f bones; bones.x = 1.0f; bones.y = 1.0f;
        #pragma unroll
        for (int i = 0; i < 8; ++i) {
            v2f a;
            a.x = lds[i * 64 + lane];
            a.y = lds[i * 64 + 32 + lane];
            acc = __builtin_amdgcn_wmma_f32_16x16x4_f32(
                /*neg_a=*/false, a, /*neg_b=*/false, bones,
                /*c_mod=*/(short)0, acc, /*reuse_a=*/false, /*reuse_b=*/false);
        }
        // lanes 0-15 hold D[M=0..7, N=lane]; lanes 16-31 hold D[M=8..15, N=lane-16]
        float tsum = acc[0] + acc[1] + acc[2] + acc[3] + acc[4] + acc[5] + acc[6] + acc[7];
        tsum += __shfl_xor(tsum, 16, 32);   // combine the two M-halves
        if (lane == 0) out[0] = tsum;
    }
}

extern "C" void kernel_launch(void* const* d_in, const int* in_sizes, int n_in,
                              void* d_out, int out_size, void* d_ws, size_t ws_size,
                              hipStream_t stream) {
    const float* x    = (const float*)d_in[0];
    const int*   pids = (const int*)d_in[1];
    const int*   vids = (const int*)d_in[2];
    float*       out  = (float*)d_out;
    float*       part = (float*)d_ws;   // BATCH*NCH*3 floats = 61,440 B

    alsr_partials<<<dim3(BATCH_SZ, NCH), 256, 0, stream>>>(x, part);
    alsr_finalize<<<1, BATCH_SZ, 0, stream>>>(x, pids, vids, part, out);
}